// CliffordLinearSimple_13950053777617
// MI455X (gfx1250) — compile-verified
//
#include <hip/hip_runtime.h>
#include <stdint.h>

// Problem constants (from reference)
#define BATCH   256
#define IN_F    1024
#define OUT_F   1024
#define D1      9
#define MV_DIM  256
#define KDIM    (IN_F * D1)    // 9216
#define NDIM    (OUT_F * D1)   // 9216

typedef __attribute__((ext_vector_type(2))) float v2f;
typedef __attribute__((ext_vector_type(8))) float v8f;
typedef uint32_t u32;
typedef u32 u32x4 __attribute__((ext_vector_type(4)));
typedef u32 u32x8 __attribute__((ext_vector_type(8)));

// GEMM tiling
#define BM   128
#define BN   128
#define BK   32
#define LDT  36                 // padded LDS row stride (floats): 16B aligned, conflict-free
#define NKC  (KDIM / BK)        // 288 k-chunks

// ---------------------------------------------------------------------------
// TDM descriptor for one W tile: 2D tensor (9216 x 9216 fp32, row stride 9216),
// tile = 32 (k, contiguous) x 128 (n rows), DMA'd to LDS with 4-DWORD padding
// after every 32 DWORDs -> LDS row stride = 36 floats (matches Bs layout).
// Issued by one wave; completion tracked with TENSORcnt.
// ---------------------------------------------------------------------------
__device__ __forceinline__ void tdm_load_w_tile(const float* gptr, u32 lds_byte_off)
{
    const uint64_t ga = (uint64_t)(uintptr_t)gptr;

    u32x4 g0;
    g0[0] = 1u;                                        // count=1, user descriptor
    g0[1] = lds_byte_off;                              // lds_addr (bytes)
    g0[2] = (u32)ga;                                   // global_addr[31:0]
    g0[3] = ((u32)(ga >> 32) & 0x01FFFFFFu)            // global_addr[56:32]
          | (2u << 30);                                // type = 2 ("image")

    u32x8 g1;
    g1[0] = (2u << 16)                                 // data_size = 4 bytes
          | (1u << 20)                                 // pad_enable
          | (4u << 22)                                 // pad_interval: after 32 DWORDs
          | (3u << 25);                                // pad_amount: 4 DWORDs
    g1[1] = ((u32)KDIM & 0xFFFFu) << 16;               // tensor_dim0[15:0]  @ bits 63:48
    g1[2] = ((u32)KDIM >> 16)                          // tensor_dim0[31:16] @ bits 79:64
          | (((u32)NDIM & 0xFFFFu) << 16);             // tensor_dim1[15:0]  @ bits 95:80
    g1[3] = ((u32)NDIM >> 16)                          // tensor_dim1[31:16] @ bits 111:96
          | ((u32)BK << 16);                           // tile_dim0 = 32     @ bits 127:112
    g1[4] = (u32)BN;                                   // tile_dim1 = 128    @ bits 143:128
    g1[5] = (u32)KDIM;                                 // tensor_dim0_stride[31:0]
    g1[6] = 0u;                                        // stride[47:32]=0, dim1_stride lo=0
    g1[7] = 0u;

    asm volatile("tensor_load_to_lds %0, %1"
                 :: "s"(g0), "s"(g1)
                 : "memory");
}

// ---------------------------------------------------------------------------
// Kernel 1: sv_out[m,n] = sum_k sv[m,k] * W[n,k] + bias[n]
//   sv[m, i*9+j] = x[m, i, (j==0 ? 0 : 1<<(j-1))]   (fused gather, manual path)
// 256 threads = 8 waves (wave32), wave grid 2(M) x 4(N), wave tile 64x32,
// each wave holds 4x2 accumulators for v_wmma_f32_16x16x4_f32.
// W tiles arrive via the Tensor Data Mover (async DMA -> LDS).
// ---------------------------------------------------------------------------
__global__ __launch_bounds__(256)
void clifford_gemm_kernel(const float* __restrict__ x,
                          const float* __restrict__ W,
                          const float* __restrict__ bias,
                          float* __restrict__ svout)
{
    __shared__ float As[2][BM][LDT];
    __shared__ float Bs[2][BN][LDT];

    const int tid   = threadIdx.x;
    const int lane  = tid & 31;
    const int wave  = tid >> 5;
    const int wm    = wave >> 2;        // 0..1  (M direction)
    const int wn    = wave & 3;         // 0..3  (N direction)
    const int lmod  = lane & 15;
    const int lhalf = lane >> 4;

    const int n0 = blockIdx.x * BN;
    const int m0 = blockIdx.y * BM;

    // A gather mapping: each thread owns one m row and 16 consecutive k's
    const int am  = tid >> 1;            // 0..127
    const int akk = (tid & 1) * 16;      // 0 or 16

    v8f acc[4][2];
#pragma unroll
    for (int tm = 0; tm < 4; ++tm)
#pragma unroll
        for (int tn = 0; tn < 2; ++tn)
            acc[tm][tn] = (v8f){0.f, 0.f, 0.f, 0.f, 0.f, 0.f, 0.f, 0.f};

    float areg[16];     // A staging: 128*32 / 256 threads = 16 floats

    const u32 bs_off0 = (u32)(uintptr_t)&Bs[0][0][0];  // LDS byte offsets
    const u32 bs_off1 = (u32)(uintptr_t)&Bs[1][0][0];

    // ---- prologue: chunk 0 --------------------------------------------------
    if (wave == 0 && lane == 0)
        tdm_load_w_tile(&W[(size_t)n0 * KDIM + 0], bs_off0);
#pragma unroll
    for (int e = 0; e < 16; ++e) {
        int k   = akk + e;
        int i   = k / D1;
        int j   = k - i * D1;
        int col = (j == 0) ? 0 : (1 << (j - 1));
        As[0][am][akk + e] = x[((size_t)(m0 + am) * IN_F + i) * MV_DIM + col];
    }
    if (wave == 0)
        __builtin_amdgcn_s_wait_tensorcnt(0);
    __syncthreads();

    // ---- main loop: issue TDM(k+1) -> gather A(k+1) -> compute(k) ----------
    for (int kc = 0; kc < NKC; ++kc) {
        const int  buf      = kc & 1;
        const bool has_next = (kc + 1 < NKC);

        if (has_next) {
            const int k0n = (kc + 1) * BK;
            if (wave == 0 && lane == 0)
                tdm_load_w_tile(&W[(size_t)n0 * KDIM + k0n],
                                (buf ^ 1) ? bs_off1 : bs_off0);
#pragma unroll
            for (int e = 0; e < 16; ++e) {
                int k   = k0n + akk + e;
                int i   = k / D1;
                int j   = k - i * D1;
                int col = (j == 0) ? 0 : (1 << (j - 1));
                areg[e] = x[((size_t)(m0 + am) * IN_F + i) * MV_DIM + col];
            }
        }

        // compute on buffer `buf` (overlaps the in-flight TDM + gather loads)
        const int mw = wm * 64;
        const int nw = wn * 32;
#pragma unroll
        for (int ks = 0; ks < BK / 4; ++ks) {
            v2f af[4], bf[2];
#pragma unroll
            for (int tm = 0; tm < 4; ++tm)
                af[tm] = *(const v2f*)&As[buf][mw + tm * 16 + lmod][ks * 4 + lhalf * 2];
#pragma unroll
            for (int tn = 0; tn < 2; ++tn)
                bf[tn] = *(const v2f*)&Bs[buf][nw + tn * 16 + lmod][ks * 4 + lhalf * 2];
#pragma unroll
            for (int tm = 0; tm < 4; ++tm)
#pragma unroll
                for (int tn = 0; tn < 2; ++tn)
                    acc[tm][tn] = __builtin_amdgcn_wmma_f32_16x16x4_f32(
                        false, af[tm], false, bf[tn],
                        (short)0, acc[tm][tn], false, false);
        }

        // commit staged A chunk k+1 to the other buffer
        if (has_next) {
            const int nb = buf ^ 1;
#pragma unroll
            for (int e = 0; e < 16; ++e)
                As[nb][am][akk + e] = areg[e];
        }

        // wave0 owns the TDM: drain TENSORcnt before releasing the barrier
        if (has_next && wave == 0)
            __builtin_amdgcn_s_wait_tensorcnt(0);
        __syncthreads();
    }

    // ---- epilogue: bias add + store sv_out ---------------------------------
    // C/D layout: VGPR i holds M = i + 8*(lane>=16), N = lane%16
    {
        const int mw = wm * 64;
        const int nw = wn * 32;
#pragma unroll
        for (int tm = 0; tm < 4; ++tm)
#pragma unroll
            for (int tn = 0; tn < 2; ++tn)
#pragma unroll
                for (int i = 0; i < 8; ++i) {
                    int row = m0 + mw + tm * 16 + lhalf * 8 + i;
                    int col = n0 + nw + tn * 16 + lmod;
                    svout[(size_t)row * NDIM + col] = acc[tm][tn][i] + bias[col];
                }
    }
}

// ---------------------------------------------------------------------------
// Kernel 2: expand sv_out (B, OUT_F, 9) -> out (B, OUT_F, 256).
// One wave per (b, out_f) row. Slot classification by popcount:
//   popc(p)==0 -> sv[0]
//   popc(p)==1 -> sv[1+ctz(p)]
//   popc(p)==2 -> sv[1+ctz(p)] * sv[1+msb(p)]     (bivector product)
//   else       -> 0
// Every lane writes 8 consecutive floats (2x float4) -> full row coverage,
// so the whole 256 MB output is written (zeros included).
// ---------------------------------------------------------------------------
__global__ __launch_bounds__(256)
void clifford_expand_kernel(const float* __restrict__ svout,
                            float* __restrict__ out)
{
    const int lane = threadIdx.x & 31;
    const int wave = threadIdx.x >> 5;
    const int row  = blockIdx.x * 8 + wave;        // 0 .. B*OUT_F-1

    const float* svp = svout + (size_t)row * D1;   // 9 contiguous floats
    float s[D1];
#pragma unroll
    for (int j = 0; j < D1; ++j) s[j] = svp[j];    // broadcast loads (L2 hit)

    float vals[8];
#pragma unroll
    for (int q = 0; q < 8; ++q) {
        const int p  = lane * 8 + q;
        const int pc = __builtin_popcount(p);
        float v = 0.0f;
        if (pc == 0) {
            v = s[0];
        } else if (pc == 1) {
            v = s[1 + __builtin_ctz(p)];
        } else if (pc == 2) {
            const int lo = __builtin_ctz(p);
            const int hi = 31 - __builtin_clz(p);
            v = s[1 + lo] * s[1 + hi];
        }
        vals[q] = v;
    }

    float* op = out + (size_t)row * MV_DIM + lane * 8;
    *(float4*)(op)     = make_float4(vals[0], vals[1], vals[2], vals[3]);
    *(float4*)(op + 4) = make_float4(vals[4], vals[5], vals[6], vals[7]);
}

// ---------------------------------------------------------------------------
extern "C" void kernel_launch(void* const* d_in, const int* in_sizes, int n_in,
                              void* d_out, int out_size, void* d_ws, size_t ws_size,
                              hipStream_t stream)
{
    const float* x    = (const float*)d_in[0];   // (256, 1024, 256)
    const float* W    = (const float*)d_in[1];   // (9216, 9216)
    const float* bias = (const float*)d_in[2];   // (9216,)
    float* out   = (float*)d_out;                // (256, 1024, 256)
    float* svout = (float*)d_ws;                 // (256, 9216) = 9.4 MB scratch

    dim3 g1(NDIM / BN, BATCH / BM);              // (72, 2)
    clifford_gemm_kernel<<<g1, 256, 0, stream>>>(x, W, bias, svout);

    const int rows = BATCH * OUT_F;              // 262144
    clifford_expand_kernel<<<rows / 8, 256, 0, stream>>>(svout, out);
}